// SOR1d_22479858827487
// MI455X (gfx1250) — compile-verified
//
#include <hip/hip_runtime.h>
#include <cstdint>

// ---------------------------------------------------------------------------
// Red-black SOR, one row (W=16384 fp32) per workgroup. Each thread holds a
// contiguous 32-point chunk of u (and the folded source term) in REGISTERS;
// a sweep needs only one LDS halo read + one publish per pass (4 DS ops per
// sweep) plus 64 independent FMAs. All bulk HBM traffic (u0 in, f in, result
// out) is moved by the CDNA5 Tensor Data Mover through a 64 KB LDS staging
// buffer; the first 4 KB of that buffer doubles as the halo slots.
// ---------------------------------------------------------------------------

#define SOR_W     16384              // points per row
#define SOR_BLK   512                // threads per block (16 wave32 waves)
#define SOR_LAST  (SOR_BLK - 1)
#define SOR_CPT   32                 // contiguous points per thread

typedef unsigned int u32x4 __attribute__((ext_vector_type(4)));
typedef int          i32x4 __attribute__((ext_vector_type(4)));
typedef int          i32x8 __attribute__((ext_vector_type(8)));

#if defined(__has_builtin)
#  if __has_builtin(__builtin_amdgcn_tensor_load_to_lds) && \
      __has_builtin(__builtin_amdgcn_tensor_store_from_lds) && \
      __has_builtin(__builtin_amdgcn_s_wait_tensorcnt)
#    define SOR_USE_TDM 1
#  endif
#endif

#ifdef SOR_USE_TDM
// TDM descriptor for a 1-D row copy of SOR_W fp32 elements.
// Group0: count=1 (valid), lds_addr, global_addr[56:0], type=2 ("image").
// Group1: data_size=2 (4B), tensor_dim0=tile_dim0=16384, tensor_dim1=tile_dim1=1,
//         tensor_dim0_stride=16384, no padding / iterate / multicast.
__device__ __forceinline__ void sor_tdm_groups(uint64_t gva, uint32_t lds_off,
                                               u32x4* g0, i32x8* g1) {
  u32x4 a;
  a.x = 1u;                                                    // count = 1
  a.y = lds_off;                                               // lds_addr
  a.z = (uint32_t)gva;                                         // global_addr[31:0]
  a.w = (((uint32_t)(gva >> 32)) & 0x01FFFFFFu) | (2u << 30);  // addr[56:32] | type=2
  *g0 = a;
  i32x8 b;
  b[0] = (int)0x00020000;  // wg_mask=0, data_size=2 (4 bytes)
  b[1] = (int)0x40000000;  // tensor_dim0.lo16 = 16384 (<<16)
  b[2] = (int)0x00010000;  // tensor_dim1.lo16 = 1 (<<16), tensor_dim0.hi16 = 0
  b[3] = (int)0x40000000;  // tile_dim0 = 16384 (<<16), tensor_dim1.hi16 = 0
  b[4] = (int)0x00000001;  // tile_dim1 = 1, tile_dim2 = 0
  b[5] = (int)SOR_W;       // tensor_dim0_stride[31:0] = 16384
  b[6] = 0;
  b[7] = 0;
  *g1 = b;
}

__device__ __forceinline__ void sor_tdm_load(const float* g, uint32_t lds_off) {
  u32x4 g0; i32x8 g1;
  sor_tdm_groups((uint64_t)(uintptr_t)g, lds_off, &g0, &g1);
  i32x4 z4 = {0, 0, 0, 0};
#if __clang_major__ >= 23
  i32x8 z8 = {0, 0, 0, 0, 0, 0, 0, 0};
  __builtin_amdgcn_tensor_load_to_lds(g0, g1, z4, z4, z8, 0);
#else
  __builtin_amdgcn_tensor_load_to_lds(g0, g1, z4, z4, 0);
#endif
}

__device__ __forceinline__ void sor_tdm_store(const float* g, uint32_t lds_off) {
  u32x4 g0; i32x8 g1;
  sor_tdm_groups((uint64_t)(uintptr_t)g, lds_off, &g0, &g1);
  i32x4 z4 = {0, 0, 0, 0};
#if __clang_major__ >= 23
  i32x8 z8 = {0, 0, 0, 0, 0, 0, 0, 0};
  __builtin_amdgcn_tensor_store_from_lds(g0, g1, z4, z4, z8, 0);
#else
  __builtin_amdgcn_tensor_store_from_lds(g0, g1, z4, z4, 0);
#endif
}
#endif  // SOR_USE_TDM

__global__ __launch_bounds__(SOR_BLK) void
sor1d_rb_kernel(const float* __restrict__ f, const float* __restrict__ dbc,
                const float* __restrict__ u0, const int* __restrict__ nitp,
                float* __restrict__ out) {
  __shared__ __align__(16) float su[SOR_W];  // 64 KB staging; su[0..1023] = halo slots
  float* eslot = su;            // eslot[t]: thread t's even u[32t]   (read by t-1)
  float* oslot = su + SOR_BLK;  // oslot[t]: thread t's odd  u[32t+31](read by t+1)

  const int b = blockIdx.x;
  const int t = threadIdx.x;
  const float* frow = f  + (size_t)b * SOR_W;
  const float* urow = u0 + (size_t)b * SOR_W;
  float*       orow = out + (size_t)b * SOR_W;

  const float half_w  = 0.5f * 1.9f;              // omega/2
  const float one_m_w = 1.0f - 1.9f;              // 1 - omega
  const float hgrid   = 1.0f / (float)(SOR_W - 1);
  const float src_c   = half_w * hgrid * hgrid;   // omega/2 * h^2

  const float bc0 = dbc[(size_t)b * SOR_W];
  const float bc1 = dbc[(size_t)b * SOR_W + SOR_W - 1];
  const int niter = nitp[0];

  // ---- stage u0 row then f row through LDS into registers (TDM DMAs) ----
  // Thread t owns contiguous points 32t .. 32t+31:
  //   local even l=2e (global j=32t+2e), local odd l=2o+1 (global j=32t+2o+1).
  float u[SOR_CPT];
  float go[SOR_CPT / 2], ge[SOR_CPT / 2];
#ifdef SOR_USE_TDM
  if (t < 32) {
    sor_tdm_load(urow, (uint32_t)(uintptr_t)su);
    __builtin_amdgcn_s_wait_tensorcnt((short)0);
  }
  __syncthreads();
  #pragma unroll
  for (int c = 0; c < SOR_CPT; ++c) u[c] = su[SOR_CPT * t + c];
  __syncthreads();
  if (t < 32) {
    sor_tdm_load(frow, (uint32_t)(uintptr_t)su);
    __builtin_amdgcn_s_wait_tensorcnt((short)0);
  }
  __syncthreads();
  #pragma unroll
  for (int o = 0; o < SOR_CPT / 2; ++o)
    go[o] = -src_c * su[SOR_CPT * t + 2 * o + 1];
  #pragma unroll
  for (int e = 0; e < SOR_CPT / 2; ++e)
    ge[e] = -src_c * su[SOR_CPT * t + 2 * e];
  __syncthreads();  // all f reads done; slot area of su may now be reused
#else
  #pragma unroll
  for (int c = 0; c < SOR_CPT; ++c) u[c] = urow[SOR_CPT * t + c];
  #pragma unroll
  for (int o = 0; o < SOR_CPT / 2; ++o)
    go[o] = -src_c * frow[SOR_CPT * t + 2 * o + 1];
  #pragma unroll
  for (int e = 0; e < SOR_CPT / 2; ++e)
    ge[e] = -src_c * frow[SOR_CPT * t + 2 * e];
#endif

  // seed the even halo with u0 values for iteration 0's odd pass
  eslot[t] = u[0];
  __syncthreads();

  // One red-black sweep: 64 independent FMAs, 2 halo loads, 2 publishes.
  auto sweep = [&]() {
    // odd (black) pass: needs right-neighbor's OLD first even at the edge
    const float haloE = eslot[(t + 1) & SOR_LAST];  // wrap read unused at t=511
    #pragma unroll
    for (int o = 0; o < SOR_CPT / 2 - 1; ++o) {
      u[2 * o + 1] = fmaf(half_w, u[2 * o] + u[2 * o + 2],
                          fmaf(one_m_w, u[2 * o + 1], go[o]));
    }
    if (t < SOR_LAST) {  // t=511,o=15 would be the j=W-1 boundary: skip
      u[SOR_CPT - 1] = fmaf(half_w, u[SOR_CPT - 2] + haloE,
                            fmaf(one_m_w, u[SOR_CPT - 1], go[SOR_CPT / 2 - 1]));
    }
    oslot[t] = u[SOR_CPT - 1];  // publish my NEW last odd for thread t+1
    __syncthreads();
    // even (red) pass: needs left-neighbor's NEW last odd at the edge
    const float haloO = oslot[(t - 1) & SOR_LAST];  // wrap read unused at t=0
    if (t > 0) {  // t=0,e=0 would be the j=0 boundary: skip
      u[0] = fmaf(half_w, haloO + u[1], fmaf(one_m_w, u[0], ge[0]));
    }
    #pragma unroll
    for (int e = 1; e < SOR_CPT / 2; ++e) {
      u[2 * e] = fmaf(half_w, u[2 * e - 1] + u[2 * e + 1],
                      fmaf(one_m_w, u[2 * e], ge[e]));
    }
    eslot[t] = u[0];  // publish my NEW first even for thread t-1
    __syncthreads();
  };

  if (niter > 0) {
    // Iteration 0 must read the ORIGINAL u0 boundary values. The sweeps never
    // modify u[0] (t=0) / u[31] (t=511), and those registers are read only by
    // their owners, so the Dirichlet BC is two local register writes, once.
    sweep();
    if (t == 0)        u[0] = bc0;
    if (t == SOR_LAST) u[SOR_CPT - 1] = bc1;
    for (int it = 1; it < niter; ++it) sweep();
  }

  // ---- write the finished row back ----
#ifdef SOR_USE_TDM
  #pragma unroll
  for (int c = 0; c < SOR_CPT; ++c) su[SOR_CPT * t + c] = u[c];
  __syncthreads();
  if (t < 32) {
    sor_tdm_store(orow, (uint32_t)(uintptr_t)su);
    __builtin_amdgcn_s_wait_tensorcnt((short)0);
  }
#else
  #pragma unroll
  for (int c = 0; c < SOR_CPT; ++c) orow[SOR_CPT * t + c] = u[c];
#endif
}

extern "C" void kernel_launch(void* const* d_in, const int* in_sizes, int n_in,
                              void* d_out, int out_size, void* d_ws,
                              size_t ws_size, hipStream_t stream) {
  const float* f   = (const float*)d_in[0];
  const float* dbc = (const float*)d_in[1];
  const float* u0  = (const float*)d_in[2];
  const int*   nit = (const int*)d_in[3];
  float* out = (float*)d_out;
  const int rows = out_size / SOR_W;  // B = 512
  sor1d_rb_kernel<<<rows, SOR_BLK, 0, stream>>>(f, dbc, u0, nit, out);
}